// EMAVectorQuantizer_1692217114978
// MI455X (gfx1250) — compile-verified
//
#include <hip/hip_runtime.h>

#define BETA  0.25f
#define NB    16
#define NC    64
#define NH    32
#define NW    32
#define HW    (NH * NW)          // 1024
#define NROWS (NB * HW)          // 16384 flattened z vectors
#define NEMB  16384              // codebook entries
#define TILE  16
#define NTILES (NEMB / TILE)     // 1024 code tiles

typedef __attribute__((ext_vector_type(2))) float v2f;
typedef __attribute__((ext_vector_type(4))) float v4f;
typedef __attribute__((ext_vector_type(8))) float v8f;

// Pointer types matching the async-LDS builtin's expected parameters:
//   param0: AS1 (global)  int4*
//   param1: AS3 (shared)  int4*
typedef __attribute__((vector_size(16))) int v4i_vs;
typedef __attribute__((address_space(1))) v4i_vs* gbl_b128_t;
typedef __attribute__((address_space(3))) v4i_vs* lds_b128_t;

#if defined(__has_builtin)
#if __has_builtin(__builtin_amdgcn_global_load_async_to_lds_b128) && \
    __has_builtin(__builtin_amdgcn_s_wait_asynccnt)
#define USE_ASYNC_LDS 1
#endif
#endif

__device__ __forceinline__ void wait_async_lds() {
#if defined(USE_ASYNC_LDS)
  __builtin_amdgcn_s_wait_asynccnt(0);
#endif
}

// ---------------------------------------------------------------------------
// Kernel 0: zero the scalar loss accumulator.
// ---------------------------------------------------------------------------
__global__ void zero_loss_kernel(float* loss) { *loss = 0.0f; }

// ---------------------------------------------------------------------------
// Kernel 1: ||E_e||^2 for every codebook row.
// ---------------------------------------------------------------------------
__global__ void enorm_kernel(const float* __restrict__ emb,
                             float* __restrict__ eNorm) {
  int e = blockIdx.x * blockDim.x + threadIdx.x;
  if (e >= NEMB) return;
  const v4f* p = (const v4f*)(emb + (size_t)e * NC);
  float s = 0.0f;
#pragma unroll
  for (int i = 0; i < NC / 4; ++i) {
    v4f v = p[i];
    s += v.x * v.x + v.y * v.y + v.z * v.z + v.w * v.w;
  }
  eNorm[e] = s;
}

// ---------------------------------------------------------------------------
// Kernel 2: fused score-GEMM + argmin.
//   score(n, e) = ||E_e||^2 - 2 * z_n . E_e   (||z_n||^2 dropped: row-const)
// 256 threads = 8 waves; wave w owns rows [blk*128 + w*16, +16).
// A fragments (16 rows x 64 dims) live in registers for all 1024 code tiles.
// B tile (16 codes x 64 dims, 4KB) staged to LDS (async, double buffered) and
// shared by all 8 waves.  16x chained V_WMMA_F32_16X16X4_F32 per tile.
// ---------------------------------------------------------------------------
__global__ __launch_bounds__(256)
void argmin_kernel(const float* __restrict__ z, const float* __restrict__ emb,
                   const float* __restrict__ eNorm,
                   unsigned* __restrict__ bestIdx) {
  __shared__ float ldsB[2][TILE * NC];  // 2 x 4 KB double buffer

  const int tid  = threadIdx.x;
  const int lane = tid & 31;
  const int wave = tid >> 5;   // 0..7
  const int half = lane >> 4;  // 0 | 1 (K sub-split for WMMA frags)
  const int sub  = lane & 15;  // row index (A) / code column (B, C/D)

  const int row0 = (blockIdx.x * 8 + wave) * TILE;
  const int bb   = row0 >> 10;        // batch
  const int hw0  = row0 & (HW - 1);   // never crosses a batch (16 | 1024)

  // ---- load A fragments once (z is BCHW: channel stride = HW) -------------
  // WMMA 16x4 f32 A layout: lanes 0-15 -> K = k0,k0+1 ; lanes 16-31 -> k0+2,k0+3
  v2f afrag[16];
  const float* zb = z + (size_t)bb * NC * HW + hw0 + sub;
#pragma unroll
  for (int kk = 0; kk < 16; ++kk) {
    const int k0 = kk * 4 + half * 2;
    afrag[kk].x = zb[(size_t)k0 * HW];
    afrag[kk].y = zb[(size_t)(k0 + 1) * HW];
  }

  float    rmin[8];
  unsigned ridx[8];
#pragma unroll
  for (int v = 0; v < 8; ++v) { rmin[v] = 3.4e38f; ridx[v] = 0u; }

  // ---- B tile staging: 256 threads x 16B = 4KB, coalesced -----------------
  auto stage = [&](int buf, int t) {
    const float* src = emb + (size_t)t * TILE * NC + tid * 4;
    float*       dst = &ldsB[buf][tid * 4];
#if defined(USE_ASYNC_LDS)
    __builtin_amdgcn_global_load_async_to_lds_b128(
        (gbl_b128_t)(unsigned long long)src,
        (lds_b128_t)(unsigned)(unsigned long long)dst, 0, 0);
#else
    *(v4f*)dst = *(const v4f*)src;
#endif
  };

  stage(0, 0);

  for (int t = 0; t < NTILES; ++t) {
    const int cur = t & 1;
    wait_async_lds();
    __syncthreads();                     // buf[cur] fully resident
    if (t + 1 < NTILES) stage(cur ^ 1, t + 1);

    const int   e0 = t * TILE;
    const float eN = eNorm[e0 + sub];    // column code's squared norm

    // B 4x16 f32 layout: lane n=sub holds code e0+sub; lanes 0-15 -> K=k0,k0+1,
    // lanes 16-31 -> K=k0+2,k0+3  => contiguous float2 in the code row.
    const float* bbase = &ldsB[cur][sub * NC + half * 2];

    v8f acc = {};
#pragma unroll
    for (int kk = 0; kk < 16; ++kk) {
      v2f bf = *(const v2f*)(bbase + kk * 4);
      acc = __builtin_amdgcn_wmma_f32_16x16x4_f32(
          false, afrag[kk], false, bf, (short)0, acc, false, false);
    }

    const unsigned code = (unsigned)(e0 + sub);
#pragma unroll
    for (int v = 0; v < 8; ++v) {
      float s = eN - 2.0f * acc[v];
      if (s < rmin[v]) { rmin[v] = s; ridx[v] = code; }
    }
    __syncthreads();                     // everyone done with buf[cur]
  }

  // ---- cross-lane argmin: row (v + 8*half) lives across a 16-lane group ---
#pragma unroll
  for (int v = 0; v < 8; ++v) {
    float    m = rmin[v];
    unsigned i = ridx[v];
#pragma unroll
    for (int off = 1; off < 16; off <<= 1) {
      float    om = __shfl_xor(m, off, 32);
      unsigned oi = (unsigned)__shfl_xor((int)i, off, 32);
      if (om < m || (om == m && oi < i)) { m = om; i = oi; }
    }
    if (sub == 0) bestIdx[row0 + v + half * 8] = i;
  }
}

// ---------------------------------------------------------------------------
// Kernel 3: gather winning codes -> BCHW output + commitment loss.
// Consecutive threads = consecutive hw -> coalesced BCHW reads/writes.
// ---------------------------------------------------------------------------
__global__ void gather_kernel(const float* __restrict__ z,
                              const float* __restrict__ emb,
                              const unsigned* __restrict__ bestIdx,
                              float* __restrict__ out,
                              float* __restrict__ loss) {
  int n = blockIdx.x * blockDim.x + threadIdx.x;
  if (n >= NROWS) return;
  const int bb = n >> 10;
  const int hw = n & (HW - 1);
  const unsigned e = bestIdx[n];
  const float* code = emb + (size_t)e * NC;
  const float* zp   = z   + (size_t)bb * NC * HW + hw;
  float*       op   = out + (size_t)bb * NC * HW + hw;
  float ls = 0.0f;
#pragma unroll
  for (int c = 0; c < NC; ++c) {
    const float q  = code[c];
    const float zc = zp[(size_t)c * HW];
    const float d  = q - zc;
    ls += d * d;
    op[(size_t)c * HW] = q;   // STE forward value == z_q
  }
  atomicAdd(loss, ls * (BETA / (float)((size_t)NROWS * NC)));
}

// ---------------------------------------------------------------------------
extern "C" void kernel_launch(void* const* d_in, const int* in_sizes, int n_in,
                              void* d_out, int out_size, void* d_ws,
                              size_t ws_size, hipStream_t stream) {
  const float* z   = (const float*)d_in[0];   // [16,64,32,32]
  const float* emb = (const float*)d_in[1];   // [16384,64]
  float* out  = (float*)d_out;                // [16,64,32,32] then loss scalar
  float* loss = out + (out_size - 1);

  float*    eNorm   = (float*)d_ws;                               // 64 KB
  unsigned* bestIdx = (unsigned*)((char*)d_ws + NEMB * sizeof(float)); // 64 KB

  zero_loss_kernel<<<1, 1, 0, stream>>>(loss);
  enorm_kernel<<<NEMB / 256, 256, 0, stream>>>(emb, eNorm);
  argmin_kernel<<<NROWS / (8 * TILE), 256, 0, stream>>>(z, emb, eNorm, bestIdx);
  gather_kernel<<<NROWS / 256, 256, 0, stream>>>(z, emb, bestIdx, out, loss);
}